// Predictor_49615462203902
// MI455X (gfx1250) — compile-verified
//
#include <hip/hip_runtime.h>
#include <math.h>

// MI455X / gfx1250, wave32. fp32 end-to-end; matrix ops via V_WMMA_F32_16X16X4_F32.

typedef __attribute__((ext_vector_type(2))) float v2f;
typedef __attribute__((ext_vector_type(8))) float v8f;

#define NB 8
#define NN 20000
#define NF 8
#define NP 12
#define NE 320000
#define BNODES (NB * NN)   // 160000
#define LDIM 64
#define NHID 38
#define NOUTC 12

__device__ __forceinline__ float sigmf(float x) { return 1.0f / (1.0f + expf(-x)); }

// ---------------- setup kernels (run once per launch) ----------------

__global__ void k_init(float* Ha, float* Hacc, int* cnt) {
  size_t i = (size_t)blockIdx.x * 256 + threadIdx.x;
  if (i < (size_t)BNODES * LDIM) { Ha[i] = 0.f; Hacc[i] = 0.f; }
  if (i < NN) cnt[i] = 0;
}

__global__ void k_deg(const int* __restrict__ dst, int* __restrict__ cnt) {
  int e = blockIdx.x * blockDim.x + threadIdx.x;
  if (e < NE) atomicAdd(&cnt[dst[e]], 1);
}

__global__ void k_dinv(const int* __restrict__ cnt, float* __restrict__ dinv,
                       float* __restrict__ self_norm) {
  int n = blockIdx.x * blockDim.x + threadIdx.x;
  if (n < NN) {
    float d = rsqrtf((float)(cnt[n] + 1));
    dinv[n] = d;
    self_norm[n] = d * d;
  }
}

__global__ void k_norm(const int* __restrict__ src, const int* __restrict__ dst,
                       const float* __restrict__ dinv, float* __restrict__ norm_e) {
  int e = blockIdx.x * blockDim.x + threadIdx.x;
  if (e < NE) norm_e[e] = dinv[src[e]] * dinv[dst[e]];
}

// single-block exclusive scan over NN degree counts -> CSR row starts
__global__ void k_scan(const int* __restrict__ cnt, int* __restrict__ rowstart,
                       int* __restrict__ fillptr) {
  __shared__ int part[256];
  const int t = threadIdx.x;
  const int CH = (NN + 255) / 256;
  const int beg = t * CH;
  const int end = (beg + CH < NN) ? (beg + CH) : NN;
  int s = 0;
  for (int n = beg; n < end; ++n) s += cnt[n];
  part[t] = s;
  __syncthreads();
  if (t == 0) {
    int run = 0;
    for (int i = 0; i < 256; ++i) { int v = part[i]; part[i] = run; run += v; }
    rowstart[NN] = run;
  }
  __syncthreads();
  int run = part[t];
  for (int n = beg; n < end; ++n) { rowstart[n] = run; fillptr[n] = run; run += cnt[n]; }
}

__global__ void k_fill(const int* __restrict__ dst, int* __restrict__ fillptr,
                       int* __restrict__ csr) {
  int e = blockIdx.x * blockDim.x + threadIdx.x;
  if (e < NE) {
    int p = atomicAdd(&fillptr[dst[e]], 1);
    csr[p] = e;
  }
}

// deterministic edge order inside each row (atomic fill order varies run-to-run)
__global__ void k_sort(const int* __restrict__ rowstart, int* __restrict__ csr) {
  int n = blockIdx.x * blockDim.x + threadIdx.x;
  if (n >= NN) return;
  const int beg = rowstart[n], end = rowstart[n + 1];
  for (int i = beg + 1; i < end; ++i) {
    int key = csr[i];
    int j = i - 1;
    while (j >= beg && csr[j] > key) { csr[j + 1] = csr[j]; --j; }
    csr[j + 1] = key;
  }
}

__global__ void k_probs(const float* __restrict__ att, float* __restrict__ probs) {
  float mx = att[0];
  for (int i = 1; i < NP; ++i) mx = fmaxf(mx, att[i]);
  float e[NP];
  float s = 0.f;
  for (int i = 0; i < NP; ++i) { e[i] = expf(att[i] - mx); s += e[i]; }
  for (int i = 0; i < NP; ++i) probs[i] = e[i] / s;
}

// ---------------- per-step kernels ----------------

// XW for the 3 gates: (160000x8)@(8x64) with two 16x16x4 f32 WMMAs per tile.
// One wave per 16-node tile; the A fragment (x tile) is reused across gates/cols.
__global__ __launch_bounds__(256) void k_xw(const float* __restrict__ xt,
    const float* __restrict__ Wz, const float* __restrict__ Wr, const float* __restrict__ Wh,
    float* __restrict__ xwZ, float* __restrict__ xwR, float* __restrict__ xwH) {
  const int lane = threadIdx.x & 31;
  const int wave = (blockIdx.x * 256 + threadIdx.x) >> 5;
  const int r0 = wave * 16;
  const int m = lane & 15;
  const int half = lane >> 4;
  const size_t rowA = (size_t)(r0 + m) * NF;
  const v2f a0 = *(const v2f*)(xt + rowA + 2 * half);      // K=0..3 chunk
  const v2f a1 = *(const v2f*)(xt + rowA + 4 + 2 * half);  // K=4..7 chunk
  const v8f vzero = {0.f, 0.f, 0.f, 0.f, 0.f, 0.f, 0.f, 0.f};
#pragma unroll
  for (int g = 0; g < 3; ++g) {
    const float* W = (g == 0) ? Wz : ((g == 1) ? Wr : Wh);
    float* o = (g == 0) ? xwZ : ((g == 1) ? xwR : xwH);
#pragma unroll
    for (int ct = 0; ct < 4; ++ct) {
      const int col = ct * 16 + m;
      v8f acc = vzero;
      v2f b0;
      b0.x = W[(2 * half) * LDIM + col];
      b0.y = W[(2 * half + 1) * LDIM + col];
      acc = __builtin_amdgcn_wmma_f32_16x16x4_f32(false, a0, false, b0, (short)0, acc, false, false);
      v2f b1;
      b1.x = W[(4 + 2 * half) * LDIM + col];
      b1.y = W[(5 + 2 * half) * LDIM + col];
      acc = __builtin_amdgcn_wmma_f32_16x16x4_f32(false, a1, false, b1, (short)0, acc, false, false);
#pragma unroll
      for (int i = 0; i < 8; ++i)
        o[(size_t)(r0 + i + 8 * half) * LDIM + col] = acc[i];
    }
  }
}

// CSR neighbor gather for all 3 gates: agg = xw*self_norm + bias + sum_e norm_e * xw[src].
// One wave per (batch,node); lane owns 2 channels. Working set sits in L2.
__global__ __launch_bounds__(256) void k_agg(
    const float* __restrict__ xwZ, const float* __restrict__ xwR, const float* __restrict__ xwH,
    const float* __restrict__ bz, const float* __restrict__ br, const float* __restrict__ bh,
    const float* __restrict__ self_norm, const float* __restrict__ norm_e,
    const int* __restrict__ rowstart, const int* __restrict__ csr, const int* __restrict__ src,
    float* __restrict__ aggZ, float* __restrict__ aggR, float* __restrict__ aggH) {
  const int lane = threadIdx.x & 31;
  const int wave = (blockIdx.x * 256 + threadIdx.x) >> 5;  // = b*NN + n
  const int b = wave / NN;
  const int n = wave - b * NN;
  const int c0 = lane * 2;
  const size_t base = (size_t)wave * LDIM + c0;
  const float sn = self_norm[n];
  v2f az = *(const v2f*)(xwZ + base) * sn;
  v2f ar = *(const v2f*)(xwR + base) * sn;
  v2f ah = *(const v2f*)(xwH + base) * sn;
  az.x += bz[c0]; az.y += bz[c0 + 1];
  ar.x += br[c0]; ar.y += br[c0 + 1];
  ah.x += bh[c0]; ah.y += bh[c0 + 1];
  const int beg = rowstart[n], end = rowstart[n + 1];
  const size_t bbase = (size_t)b * NN;
  for (int j = beg; j < end; ++j) {
    const int e = csr[j];
    const float wgt = norm_e[e];
    const size_t sb = (bbase + src[e]) * LDIM + c0;
    if (j + 1 < end) {  // gfx1250 global_prefetch_b8 for the next gathered row
      const size_t sb2 = (bbase + src[csr[j + 1]]) * LDIM + c0;
      __builtin_prefetch(xwZ + sb2, 0, 3);
      __builtin_prefetch(xwR + sb2, 0, 3);
      __builtin_prefetch(xwH + sb2, 0, 3);
    }
    az += *(const v2f*)(xwZ + sb) * wgt;
    ar += *(const v2f*)(xwR + sb) * wgt;
    ah += *(const v2f*)(xwH + sb) * wgt;
  }
  *(v2f*)(aggZ + base) = az;
  *(v2f*)(aggR + base) = ar;
  *(v2f*)(aggH + base) = ah;
}

// Fused GRU step: Z, R, Hc GEMMs (16x128 @ 128x64 each, 32 WMMAs per 16x16 tile)
// + activations + Hn = Z*H + (1-Z)*Hc + attention accumulation. One wave per 16 nodes.
// H*R is re-laid-out A-fragment-wise through padded LDS (stride 66 -> conflict-free b64).
__global__ __launch_bounds__(256) void k_gate(
    const float* __restrict__ aggZ, const float* __restrict__ aggR, const float* __restrict__ aggH,
    const float* __restrict__ Hprev, float* __restrict__ Hnext, float* __restrict__ Hacc,
    const float* __restrict__ Lzw, const float* __restrict__ Lzb,
    const float* __restrict__ Lrw, const float* __restrict__ Lrb,
    const float* __restrict__ Lhw, const float* __restrict__ Lhb,
    const float* __restrict__ probs, int t) {
  __shared__ __align__(16) float sHR[8][16 * 66];
  const int lane = threadIdx.x & 31;
  const int wsl = threadIdx.x >> 5;
  const int wave = blockIdx.x * 8 + wsl;
  const int r0 = wave * 16;
  const int m = lane & 15;
  const int half = lane >> 4;
  const size_t rowA = (size_t)(r0 + m) * LDIM;
  const float pt = probs[t];
  const v8f vzero = {0.f, 0.f, 0.f, 0.f, 0.f, 0.f, 0.f, 0.f};

  // H tile in D-fragment layout (row = i + 8*half, col = ct*16 + m)
  v8f hfrag[4];
#pragma unroll
  for (int ct = 0; ct < 4; ++ct)
#pragma unroll
    for (int i = 0; i < 8; ++i)
      hfrag[ct][i] = Hprev[(size_t)(r0 + i + 8 * half) * LDIM + ct * 16 + m];

  v8f acc[4];

  // ---------------- update gate Z ----------------
#pragma unroll
  for (int ct = 0; ct < 4; ++ct) acc[ct] = vzero;
#pragma unroll 4
  for (int kc = 0; kc < 16; ++kc) {  // K = 0..63 : agg half
    const int kk = kc * 4 + 2 * half;
    const v2f a = *(const v2f*)(aggZ + rowA + kk);
#pragma unroll
    for (int ct = 0; ct < 4; ++ct) {
      v2f bb;
      bb.x = Lzw[(size_t)kk * LDIM + ct * 16 + m];
      bb.y = Lzw[(size_t)(kk + 1) * LDIM + ct * 16 + m];
      acc[ct] = __builtin_amdgcn_wmma_f32_16x16x4_f32(false, a, false, bb, (short)0, acc[ct], false, false);
    }
  }
#pragma unroll 4
  for (int kc = 0; kc < 16; ++kc) {  // K = 64..127 : H half
    const int kk = kc * 4 + 2 * half;
    const v2f a = *(const v2f*)(Hprev + rowA + kk);
#pragma unroll
    for (int ct = 0; ct < 4; ++ct) {
      v2f bb;
      bb.x = Lzw[(size_t)(64 + kk) * LDIM + ct * 16 + m];
      bb.y = Lzw[(size_t)(65 + kk) * LDIM + ct * 16 + m];
      acc[ct] = __builtin_amdgcn_wmma_f32_16x16x4_f32(false, a, false, bb, (short)0, acc[ct], false, false);
    }
  }
  v8f zfrag[4];
#pragma unroll
  for (int ct = 0; ct < 4; ++ct)
#pragma unroll
    for (int i = 0; i < 8; ++i)
      zfrag[ct][i] = sigmf(acc[ct][i] + Lzb[ct * 16 + m]);

  // ---------------- reset gate R (consumed immediately as H*R in LDS) ----------------
#pragma unroll
  for (int ct = 0; ct < 4; ++ct) acc[ct] = vzero;
#pragma unroll 4
  for (int kc = 0; kc < 16; ++kc) {
    const int kk = kc * 4 + 2 * half;
    const v2f a = *(const v2f*)(aggR + rowA + kk);
#pragma unroll
    for (int ct = 0; ct < 4; ++ct) {
      v2f bb;
      bb.x = Lrw[(size_t)kk * LDIM + ct * 16 + m];
      bb.y = Lrw[(size_t)(kk + 1) * LDIM + ct * 16 + m];
      acc[ct] = __builtin_amdgcn_wmma_f32_16x16x4_f32(false, a, false, bb, (short)0, acc[ct], false, false);
    }
  }
#pragma unroll 4
  for (int kc = 0; kc < 16; ++kc) {
    const int kk = kc * 4 + 2 * half;
    const v2f a = *(const v2f*)(Hprev + rowA + kk);
#pragma unroll
    for (int ct = 0; ct < 4; ++ct) {
      v2f bb;
      bb.x = Lrw[(size_t)(64 + kk) * LDIM + ct * 16 + m];
      bb.y = Lrw[(size_t)(65 + kk) * LDIM + ct * 16 + m];
      acc[ct] = __builtin_amdgcn_wmma_f32_16x16x4_f32(false, a, false, bb, (short)0, acc[ct], false, false);
    }
  }
#pragma unroll
  for (int ct = 0; ct < 4; ++ct)
#pragma unroll
    for (int i = 0; i < 8; ++i) {
      const float r = sigmf(acc[ct][i] + Lrb[ct * 16 + m]);
      sHR[wsl][(i + 8 * half) * 66 + ct * 16 + m] = hfrag[ct][i] * r;
    }
  __syncthreads();

  // ---------------- candidate Hc ----------------
#pragma unroll
  for (int ct = 0; ct < 4; ++ct) acc[ct] = vzero;
#pragma unroll 4
  for (int kc = 0; kc < 16; ++kc) {
    const int kk = kc * 4 + 2 * half;
    const v2f a = *(const v2f*)(aggH + rowA + kk);
#pragma unroll
    for (int ct = 0; ct < 4; ++ct) {
      v2f bb;
      bb.x = Lhw[(size_t)kk * LDIM + ct * 16 + m];
      bb.y = Lhw[(size_t)(kk + 1) * LDIM + ct * 16 + m];
      acc[ct] = __builtin_amdgcn_wmma_f32_16x16x4_f32(false, a, false, bb, (short)0, acc[ct], false, false);
    }
  }
#pragma unroll 4
  for (int kc = 0; kc < 16; ++kc) {
    const int kk = kc * 4 + 2 * half;
    const v2f a = *(const v2f*)(&sHR[wsl][m * 66 + kk]);  // ds_load_b64, conflict-free
#pragma unroll
    for (int ct = 0; ct < 4; ++ct) {
      v2f bb;
      bb.x = Lhw[(size_t)(64 + kk) * LDIM + ct * 16 + m];
      bb.y = Lhw[(size_t)(65 + kk) * LDIM + ct * 16 + m];
      acc[ct] = __builtin_amdgcn_wmma_f32_16x16x4_f32(false, a, false, bb, (short)0, acc[ct], false, false);
    }
  }

  // ---------------- Hn = Z*H + (1-Z)*Hc, plus attention accumulation ----------------
#pragma unroll
  for (int ct = 0; ct < 4; ++ct)
#pragma unroll
    for (int i = 0; i < 8; ++i) {
      const float hc = tanhf(acc[ct][i] + Lhb[ct * 16 + m]);
      const float z = zfrag[ct][i];
      const float hnv = z * hfrag[ct][i] + (1.f - z) * hc;
      const size_t idx = (size_t)(r0 + i + 8 * half) * LDIM + ct * 16 + m;
      Hnext[idx] = hnv;
      Hacc[idx] += pt * hnv;
    }
}

// ---------------- epilogue kernels ----------------

// deterministic two-stage per-channel sum / sum-of-squares
__global__ __launch_bounds__(256) void k_stats(const float* __restrict__ Hacc,
                                               float* __restrict__ spart) {
  __shared__ float ps[256], pq[256];
  const int tid = threadIdx.x;
  const size_t idx0 = (size_t)blockIdx.x * 256 + tid;
  const size_t stride = (size_t)gridDim.x * 256;  // multiple of 64 -> channel fixed per thread
  float s = 0.f, q = 0.f;
  for (size_t i = idx0; i < (size_t)BNODES * LDIM; i += stride) {
    const float v = Hacc[i];
    s += v;
    q += v * v;
  }
  ps[tid] = s;
  pq[tid] = q;
  __syncthreads();
  if (tid < 64) {
    const float ss = ps[tid] + ps[tid + 64] + ps[tid + 128] + ps[tid + 192];
    const float qq = pq[tid] + pq[tid + 64] + pq[tid + 128] + pq[tid + 192];
    spart[(size_t)blockIdx.x * 128 + tid] = ss;
    spart[(size_t)blockIdx.x * 128 + 64 + tid] = qq;
  }
}

__global__ void k_stats2(const float* __restrict__ spart, float* __restrict__ stats, int nblk) {
  const int tid = threadIdx.x;  // 128 threads
  float s = 0.f;
  for (int b = 0; b < nblk; ++b) s += spart[(size_t)b * 128 + tid];
  stats[tid] = s;
}

// normalize -> sigmoid -> relu(H@W1+b1)@W2+b2 -> sigmoid
__global__ __launch_bounds__(128) void k_mlp(
    const float* __restrict__ Hacc, const float* __restrict__ stats,
    const float* __restrict__ gamma, const float* __restrict__ beta,
    const float* __restrict__ W1, const float* __restrict__ b1,
    const float* __restrict__ W2, const float* __restrict__ b2,
    float* __restrict__ out) {
  __shared__ float sW1[LDIM * NHID];
  __shared__ float sW2[NHID * NOUTC];
  __shared__ float sScale[LDIM], sShift[LDIM], sB1[NHID], sB2[NOUTC];
  for (int i = threadIdx.x; i < LDIM * NHID; i += 128) sW1[i] = W1[i];
  for (int i = threadIdx.x; i < NHID * NOUTC; i += 128) sW2[i] = W2[i];
  if (threadIdx.x < LDIM) {
    const int c = threadIdx.x;
    const float invn = 1.0f / (float)BNODES;
    const float mean = stats[c] * invn;
    const float var = stats[64 + c] * invn - mean * mean;
    const float inv = rsqrtf(var + 1e-5f);
    sScale[c] = inv * gamma[c];
    sShift[c] = beta[c] - mean * inv * gamma[c];
  }
  if (threadIdx.x < NHID) sB1[threadIdx.x] = b1[threadIdx.x];
  if (threadIdx.x < NOUTC) sB2[threadIdx.x] = b2[threadIdx.x];
  __syncthreads();
  const int node = blockIdx.x * 128 + threadIdx.x;
  if (node >= BNODES) return;
  float sv[LDIM];
#pragma unroll
  for (int c = 0; c < LDIM; ++c) {
    const float h = Hacc[(size_t)node * LDIM + c];
    sv[c] = sigmf(h * sScale[c] + sShift[c]);
  }
  float o[NOUTC];
#pragma unroll
  for (int j = 0; j < NOUTC; ++j) o[j] = sB2[j];
  for (int hj = 0; hj < NHID; ++hj) {
    float a = sB1[hj];
#pragma unroll
    for (int c = 0; c < LDIM; ++c) a += sv[c] * sW1[c * NHID + hj];
    a = fmaxf(a, 0.f);
#pragma unroll
    for (int j = 0; j < NOUTC; ++j) o[j] += a * sW2[hj * NOUTC + j];
  }
#pragma unroll
  for (int j = 0; j < NOUTC; ++j) out[(size_t)node * NOUTC + j] = sigmf(o[j]);
}

// ---------------- host orchestration ----------------

extern "C" void kernel_launch(void* const* d_in, const int* in_sizes, int n_in,
                              void* d_out, int out_size, void* d_ws, size_t ws_size,
                              hipStream_t stream) {
  const float* x     = (const float*)d_in[0];
  const int*   ei    = (const int*)d_in[1];
  const float* att   = (const float*)d_in[3];
  const float* Wz    = (const float*)d_in[4];
  const float* bz    = (const float*)d_in[5];
  const float* Lzw   = (const float*)d_in[6];
  const float* Lzb   = (const float*)d_in[7];
  const float* Wr    = (const float*)d_in[8];
  const float* br    = (const float*)d_in[9];
  const float* Lrw   = (const float*)d_in[10];
  const float* Lrb   = (const float*)d_in[11];
  const float* Wh    = (const float*)d_in[12];
  const float* bh    = (const float*)d_in[13];
  const float* Lhw   = (const float*)d_in[14];
  const float* Lhb   = (const float*)d_in[15];
  const float* gamma = (const float*)d_in[16];
  const float* beta  = (const float*)d_in[17];
  const float* W1    = (const float*)d_in[18];
  const float* b1    = (const float*)d_in[19];
  const float* W2    = (const float*)d_in[20];
  const float* b2    = (const float*)d_in[21];
  float* out = (float*)d_out;

  const int* srcIdx = ei;                    // reference uses only the first E edges
  const int* dstIdx = ei + (size_t)NB * NE;  // row 1 of edge_index

  char* w = (char*)d_ws;
  auto alloc = [&](size_t bytes) {
    char* p = w;
    w += (bytes + 255) & ~(size_t)255;
    return p;
  };
  int*   cnt      = (int*)alloc((size_t)NN * 4);
  int*   rowstart = (int*)alloc((size_t)(NN + 1) * 4);
  int*   fillptr  = (int*)alloc((size_t)NN * 4);
  int*   csr      = (int*)alloc((size_t)NE * 4);
  float* dinv     = (float*)alloc((size_t)NN * 4);
  float* self_n   = (float*)alloc((size_t)NN * 4);
  float* norm_e   = (float*)alloc((size_t)NE * 4);
  float* probs    = (float*)alloc(64 * 4);
  float* stats    = (float*)alloc(128 * 4);
  float* spart    = (float*)alloc((size_t)512 * 128 * 4);
  const size_t msz = (size_t)BNODES * LDIM * 4;
  float* xwZ  = (float*)alloc(msz);
  float* xwR  = (float*)alloc(msz);
  float* xwH  = (float*)alloc(msz);
  float* aggZ = (float*)alloc(msz);
  float* aggR = (float*)alloc(msz);
  float* aggH = (float*)alloc(msz);
  float* Ha   = (float*)alloc(msz);
  float* Hb   = (float*)alloc(msz);
  float* Hacc = (float*)alloc(msz);

  k_init<<<(BNODES * LDIM) / 256, 256, 0, stream>>>(Ha, Hacc, cnt);
  k_deg<<<(NE + 255) / 256, 256, 0, stream>>>(dstIdx, cnt);
  k_dinv<<<(NN + 255) / 256, 256, 0, stream>>>(cnt, dinv, self_n);
  k_norm<<<(NE + 255) / 256, 256, 0, stream>>>(srcIdx, dstIdx, dinv, norm_e);
  k_scan<<<1, 256, 0, stream>>>(cnt, rowstart, fillptr);
  k_fill<<<(NE + 255) / 256, 256, 0, stream>>>(dstIdx, fillptr, csr);
  k_sort<<<(NN + 255) / 256, 256, 0, stream>>>(rowstart, csr);
  k_probs<<<1, 1, 0, stream>>>(att, probs);

  float* hp = Ha;
  float* hn = Hb;
  for (int t = 0; t < NP; ++t) {
    k_xw<<<BNODES / (16 * 8), 256, 0, stream>>>(x + (size_t)t * BNODES * NF, Wz, Wr, Wh,
                                                xwZ, xwR, xwH);
    k_agg<<<BNODES / 8, 256, 0, stream>>>(xwZ, xwR, xwH, bz, br, bh, self_n, norm_e,
                                          rowstart, csr, srcIdx, aggZ, aggR, aggH);
    k_gate<<<BNODES / (16 * 8), 256, 0, stream>>>(aggZ, aggR, aggH, hp, hn, Hacc,
                                                  Lzw, Lzb, Lrw, Lrb, Lhw, Lhb, probs, t);
    float* tmp = hp; hp = hn; hn = tmp;
  }

  k_stats<<<512, 256, 0, stream>>>(Hacc, spart);
  k_stats2<<<1, 128, 0, stream>>>(spart, stats, 512);
  k_mlp<<<BNODES / 128, 128, 0, stream>>>(Hacc, stats, gamma, beta, W1, b1, W2, b2, out);

  (void)in_sizes; (void)n_in; (void)out_size; (void)ws_size;
}